// SplitParallelQCNN_90108413870182
// MI455X (gfx1250) — compile-verified
//
#include <hip/hip_runtime.h>
#include <math.h>

#define QPS     4
#define NSPLIT  4
#define NLAYER  4
#define DIM     16
#define NCLASS  5
#define BATCH   32768
#define FEAT    16
#define H1      128
#define H2      64

#define WAVES_PER_BLOCK 4
#define H1P 132   // 128 + 4 pad words -> bank-conflict-free (stride mod 64 = 4)
#define H2P 68    // 64 + 4 pad words

typedef __attribute__((ext_vector_type(2))) float v2f;
typedef __attribute__((ext_vector_type(8))) float v8f;

// ---------------------------------------------------------------------------
// Kernel 1: fuse per-(split,layer,qubit) rotations into one 2x2 complex U.
//   U = RZ(hz) @ RY(hy) @ RX(hx), half-angles. 64 gates x 8 floats.
// ---------------------------------------------------------------------------
__global__ void qcnn_unitaries(const float* __restrict__ sw, float* __restrict__ U) {
    int g = threadIdx.x;
    if (g >= NSPLIT * NLAYER * QPS) return;
    const float* w = sw + g * 3;   // [S,L,Q,3] row-major == gate-major
    float hx = 0.5f * w[0], hy = 0.5f * w[1], hz = 0.5f * w[2];
    float sx, cx, sy, cy, sz, cz;
    sincosf(hx, &sx, &cx);
    sincosf(hy, &sy, &cy);
    sincosf(hz, &sz, &cz);
    // M = RY @ RX
    float m00r = cy * cx, m00i =  sy * sx;
    float m01r = -sy * cx, m01i = -cy * sx;
    float m10r =  sy * cx, m10i = -cy * sx;
    float m11r =  cy * cx, m11i = -sy * sx;
    // U = diag(e^{-i hz}, e^{+i hz}) @ M
    float* o = U + g * 8;
    o[0] = cz * m00r + sz * m00i;  o[1] = cz * m00i - sz * m00r;   // u00
    o[2] = cz * m01r + sz * m01i;  o[3] = cz * m01i - sz * m01r;   // u01
    o[4] = cz * m10r - sz * m10i;  o[5] = cz * m10i + sz * m10r;   // u10
    o[6] = cz * m11r - sz * m11i;  o[7] = cz * m11i + sz * m11r;   // u11
}

// ---------------------------------------------------------------------------
// Kernel 2: one thread per (sample, split). Register-resident 16-amp
// complex statevector, fully unrolled gate network, PauliZ expectations.
// Thread t -> b = t/4, s = t%4; input chunk = x + 16*t (coalesced).
// ---------------------------------------------------------------------------
__global__ __launch_bounds__(256) void qcnn_features(const float* __restrict__ x,
                                                     const float* __restrict__ Utab,
                                                     float* __restrict__ feats) {
    __shared__ float uS[NSPLIT * NLAYER * QPS * 8];   // 2 KB
    for (int i = threadIdx.x; i < NSPLIT * NLAYER * QPS * 8; i += blockDim.x)
        uS[i] = Utab[i];
    __syncthreads();

    int t = blockIdx.x * blockDim.x + threadIdx.x;    // 0 .. B*4-1
    int s = t & 3;

    const float* xp = x + (size_t)t * DIM;
    float re[DIM], im[DIM];
    float nrm2 = 0.f;
#pragma unroll
    for (int i = 0; i < DIM; i++) { float v = xp[i]; re[i] = v; nrm2 += v * v; }
    float nrm = sqrtf(nrm2);
    float scale = (nrm > 0.f) ? (1.f / nrm) : 1.f;
#pragma unroll
    for (int i = 0; i < DIM; i++) { re[i] *= scale; im[i] = 0.f; }

    const float* us = &uS[s * NLAYER * QPS * 8];
#pragma unroll
    for (int l = 0; l < NLAYER; l++) {
#pragma unroll
        for (int j = 0; j < QPS; j++) {
            const float* u = &us[(l * QPS + j) * 8];
            float u00r = u[0], u00i = u[1], u01r = u[2], u01i = u[3];
            float u10r = u[4], u10i = u[5], u11r = u[6], u11i = u[7];
            const int m = 8 >> j;          // wire j bit (wire 0 = MSB)
#pragma unroll
            for (int idx = 0; idx < DIM; idx++) {
                if (idx & m) continue;
                const int i1 = idx | m;
                float a0r = re[idx], a0i = im[idx], a1r = re[i1], a1i = im[i1];
                re[idx] = u00r * a0r - u00i * a0i + u01r * a1r - u01i * a1i;
                im[idx] = u00r * a0i + u00i * a0r + u01r * a1i + u01i * a1r;
                re[i1]  = u10r * a0r - u10i * a0i + u11r * a1r - u11i * a1i;
                im[i1]  = u10r * a0i + u10i * a0r + u11r * a1i + u11i * a1r;
            }
        }
#pragma unroll
        for (int j = 0; j < QPS; j++) {    // CNOT ring
            const int mc = 8 >> j, mt = 8 >> ((j + 1) & 3);
#pragma unroll
            for (int idx = 0; idx < DIM; idx++) {
                if ((idx & mc) && !(idx & mt)) {
                    const int i1 = idx | mt;
                    float tr = re[idx], ti = im[idx];
                    re[idx] = re[i1]; im[idx] = im[i1];
                    re[i1] = tr;      im[i1] = ti;
                }
            }
        }
    }

    float e[QPS];
#pragma unroll
    for (int q = 0; q < QPS; q++) e[q] = 0.f;
#pragma unroll
    for (int idx = 0; idx < DIM; idx++) {
        float p = re[idx] * re[idx] + im[idx] * im[idx];
#pragma unroll
        for (int q = 0; q < QPS; q++)
            e[q] += ((idx >> (3 - q)) & 1) ? -p : p;
    }
    int b = t >> 2;
    float* f = feats + (size_t)b * FEAT + s * QPS;
#pragma unroll
    for (int q = 0; q < QPS; q++) f[q] = e[q];
}

// ---------------------------------------------------------------------------
// Kernel 3: MLP via V_WMMA_F32_16X16X4_F32. One wave per 16-row block.
// ---------------------------------------------------------------------------
__global__ __launch_bounds__(32 * WAVES_PER_BLOCK) void qcnn_mlp(
    const float* __restrict__ feats,
    const float* __restrict__ W1, const float* __restrict__ b1,
    const float* __restrict__ W2, const float* __restrict__ b2,
    const float* __restrict__ W3, const float* __restrict__ b3,
    float* __restrict__ out)
{
    __shared__ float h1s[WAVES_PER_BLOCK][16 * H1P];   // 33 KB
    __shared__ float h2s[WAVES_PER_BLOCK][16 * H2P];   // 17 KB

    const int lane = threadIdx.x & 31;
    const int wave = threadIdx.x >> 5;
    const int half = lane >> 4;      // 0 | 1
    const int nl   = lane & 15;      // sub-tile row/col index
    const int rowBase = (blockIdx.x * WAVES_PER_BLOCK + wave) * 16;
    float* h1 = h1s[wave];
    float* h2 = h2s[wave];

    // ---- Layer 1: feats[16x16] @ W1[16x128] + b1, relu -> h1 (LDS) ----
    v2f a1f[4];
#pragma unroll
    for (int kc = 0; kc < 4; kc++) {
        const float* ap = feats + (size_t)(rowBase + nl) * FEAT + kc * 4 + 2 * half;
        a1f[kc] = v2f{ap[0], ap[1]};
    }
#pragma unroll
    for (int nt = 0; nt < 8; nt++) {
        const int n0 = nt * 16;
        float bias = b1[n0 + nl];
        v8f c;
#pragma unroll
        for (int r = 0; r < 8; r++) c[r] = bias;
#pragma unroll
        for (int kc = 0; kc < 4; kc++) {
            const float* bp = W1 + (size_t)(kc * 4 + 2 * half) * H1 + n0 + nl;
            v2f bf = v2f{bp[0], bp[H1]};
            c = __builtin_amdgcn_wmma_f32_16x16x4_f32(false, a1f[kc], false, bf,
                                                      (short)0, c, false, false);
        }
#pragma unroll
        for (int r = 0; r < 8; r++)
            h1[(r + 8 * half) * H1P + n0 + nl] = fmaxf(c[r], 0.f);
    }
    __syncthreads();

    // ---- Layer 2: h1[16x128] @ W2[128x64] + b2, relu -> h2 (LDS) ----
#pragma unroll
    for (int nt = 0; nt < 4; nt++) {
        const int n0 = nt * 16;
        float bias = b2[n0 + nl];
        v8f c;
#pragma unroll
        for (int r = 0; r < 8; r++) c[r] = bias;
#pragma unroll 4
        for (int k0 = 0; k0 < H1; k0 += 4) {
            const float* ap = h1 + nl * H1P + k0 + 2 * half;
            v2f af = v2f{ap[0], ap[1]};
            const float* bp = W2 + (size_t)(k0 + 2 * half) * H2 + n0 + nl;
            v2f bf = v2f{bp[0], bp[H2]};
            c = __builtin_amdgcn_wmma_f32_16x16x4_f32(false, af, false, bf,
                                                      (short)0, c, false, false);
        }
#pragma unroll
        for (int r = 0; r < 8; r++)
            h2[(r + 8 * half) * H2P + n0 + nl] = fmaxf(c[r], 0.f);
    }
    __syncthreads();

    // ---- Layer 3: h2[16x64] @ W3[64x5] + b3 (N padded to 16, branch-free) ----
    {
        const float msk = (nl < NCLASS) ? 1.f : 0.f;
        const int ncl = (nl < NCLASS) ? nl : (NCLASS - 1);  // clamped, no OOB
        float bias = msk * b3[ncl];
        v8f c;
#pragma unroll
        for (int r = 0; r < 8; r++) c[r] = bias;
#pragma unroll 4
        for (int k0 = 0; k0 < H2; k0 += 4) {
            const float* ap = h2 + nl * H2P + k0 + 2 * half;
            v2f af = v2f{ap[0], ap[1]};
            const int k = k0 + 2 * half;
            v2f bf = v2f{msk * W3[k * NCLASS + ncl], msk * W3[(k + 1) * NCLASS + ncl]};
            c = __builtin_amdgcn_wmma_f32_16x16x4_f32(false, af, false, bf,
                                                      (short)0, c, false, false);
        }
        if (nl < NCLASS) {
#pragma unroll
            for (int r = 0; r < 8; r++) {
                const int m = r + 8 * half;
                out[(size_t)(rowBase + m) * NCLASS + nl] = c[r];
            }
        }
    }
}

// ---------------------------------------------------------------------------
extern "C" void kernel_launch(void* const* d_in, const int* in_sizes, int n_in,
                              void* d_out, int out_size, void* d_ws, size_t ws_size,
                              hipStream_t stream) {
    const float* x  = (const float*)d_in[0];
    const float* sw = (const float*)d_in[1];
    const float* W1 = (const float*)d_in[2];
    const float* b1 = (const float*)d_in[3];
    const float* W2 = (const float*)d_in[4];
    const float* b2 = (const float*)d_in[5];
    const float* W3 = (const float*)d_in[6];
    const float* b3 = (const float*)d_in[7];
    float* out = (float*)d_out;

    float* Utab  = (float*)d_ws;                         // 64 * 8 floats
    float* feats = (float*)((char*)d_ws + 4096);         // B * 16 floats (2 MB)

    qcnn_unitaries<<<dim3(1), dim3(64), 0, stream>>>(sw, Utab);

    qcnn_features<<<dim3((BATCH * NSPLIT) / 256), dim3(256), 0, stream>>>(x, Utab, feats);

    qcnn_mlp<<<dim3(BATCH / (16 * WAVES_PER_BLOCK)), dim3(32 * WAVES_PER_BLOCK), 0, stream>>>(
        feats, W1, b1, W2, b2, W3, b3, out);
}